// GATEncoder_31190052503739
// MI455X (gfx1250) — compile-verified
//
#include <hip/hip_runtime.h>
#include <hip/hip_bf16.h>

typedef float v2f __attribute__((ext_vector_type(2)));
typedef float v4f __attribute__((ext_vector_type(4)));
typedef float v8f __attribute__((ext_vector_type(8)));
typedef unsigned u4u __attribute__((ext_vector_type(4)));

// ---------- monotonic float <-> uint mapping for atomicMax on floats ----------
__device__ __forceinline__ unsigned f2o(float f) {
    unsigned u = __float_as_uint(f);
    return (u >> 31) ? ~u : (u | 0x80000000u);
}
__device__ __forceinline__ float o2f(unsigned o) {
    return (o >> 31) ? __uint_as_float(o & 0x7FFFFFFFu) : __uint_as_float(~o);
}

// ---------- vectorized fill (all our buffer sizes are multiples of 4 dwords) ----------
__global__ void fill_u32x4_k(unsigned* __restrict__ p, unsigned v, long n4) {
    long stride = (long)gridDim.x * blockDim.x;
    u4u val = {v, v, v, v};
    for (long i = (long)blockIdx.x * blockDim.x + threadIdx.x; i < n4; i += stride)
        ((u4u*)p)[i] = val;
}

// ---------- fp32 WMMA GEMM: C[M,NC] = A[M,K] @ B[K,NC] ----------
// One wave32 computes a 16x64 strip = 4 accumulators of V_WMMA_F32_16X16X4_F32,
// reusing the A fragment (one global_load_b64 per K-step) across 4 WMMAs.
// A frag (16x4): lanes 0-15 row M=l, K=k..k+1; lanes 16-31 row M=l-16, K=k+2..k+3.
// B frag (4x16): lanes 0-15 col N=l, rows k..k+1 in v0/v1; lanes 16-31 rows k+2..k+3.
// C/D (16x16):  vgpr r -> row r (lanes 0-15) / row r+8 (lanes 16-31), col = lane%16.
__global__ void wmma_gemm_f32_k(const float* __restrict__ A, const float* __restrict__ B,
                                float* __restrict__ C, int M, int K, int NC) {
    int gwave = (blockIdx.x * blockDim.x + threadIdx.x) >> 5;
    int lane  = threadIdx.x & 31;
    int Ng = NC >> 6;                       // 64-wide column groups
    int groups = (M >> 4) * Ng;
    if (gwave >= groups) return;
    int tm = gwave / Ng;
    int tg = gwave - tm * Ng;
    int row0 = tm << 4, col0 = tg << 6;
    int lrow  = lane & 15;
    int khalf = (lane >> 4) << 1;           // 0 or 2
    v8f z = {0.f, 0.f, 0.f, 0.f, 0.f, 0.f, 0.f, 0.f};
    v8f acc0 = z, acc1 = z, acc2 = z, acc3 = z;
    const float* arow  = A + (size_t)(row0 + lrow) * K + khalf;   // 8B aligned
    const float* bbase = B + col0 + lrow;
    for (int k = 0; k < K; k += 4) {
        v2f a = *(const v2f*)(arow + k);
        const float* bp0 = bbase + (size_t)(k + khalf) * NC;
        const float* bp1 = bp0 + NC;
        v2f b0, b1, b2, b3;
        b0[0] = bp0[0];  b0[1] = bp1[0];
        b1[0] = bp0[16]; b1[1] = bp1[16];
        b2[0] = bp0[32]; b2[1] = bp1[32];
        b3[0] = bp0[48]; b3[1] = bp1[48];
        acc0 = __builtin_amdgcn_wmma_f32_16x16x4_f32(false, a, false, b0, (short)0, acc0, false, false);
        acc1 = __builtin_amdgcn_wmma_f32_16x16x4_f32(false, a, false, b1, (short)0, acc1, false, false);
        acc2 = __builtin_amdgcn_wmma_f32_16x16x4_f32(false, a, false, b2, (short)0, acc2, false, false);
        acc3 = __builtin_amdgcn_wmma_f32_16x16x4_f32(false, a, false, b3, (short)0, acc3, false, false);
    }
    float* cp = C + (size_t)(row0 + (lane >> 4) * 8) * NC + col0 + lrow;
#pragma unroll
    for (int r = 0; r < 8; ++r) {
        size_t ro = (size_t)r * NC;
        cp[ro]      = acc0[r];
        cp[ro + 16] = acc1[r];
        cp[ro + 32] = acc2[r];
        cp[ro + 48] = acc3[r];
    }
}

// ---------- per-(node,head) attention coefficients: wave32 dot products ----------
__global__ void attn_coef_k(const float* __restrict__ h, const float* __restrict__ a_src,
                            const float* __restrict__ a_dst, float* __restrict__ als,
                            float* __restrict__ ald, int N, int heads, int C) {
    int wave = (blockIdx.x * blockDim.x + threadIdx.x) >> 5;
    int lane = threadIdx.x & 31;
    int total = N * heads;
    if (wave >= total) return;
    int n = wave / heads, hd = wave - n * heads;
    const float* hp = h + (size_t)n * heads * C + (size_t)hd * C;
    const float* as = a_src + (size_t)hd * C;
    const float* ad = a_dst + (size_t)hd * C;
    float s = 0.f, d = 0.f;
    for (int c = lane * 4; c < C; c += 128) {     // C is a multiple of 128
        v4f hv = *(const v4f*)(hp + c);
        v4f av = *(const v4f*)(as + c);
        v4f dv = *(const v4f*)(ad + c);
        s += hv[0] * av[0] + hv[1] * av[1] + hv[2] * av[2] + hv[3] * av[3];
        d += hv[0] * dv[0] + hv[1] * dv[1] + hv[2] * dv[2] + hv[3] * dv[3];
    }
#pragma unroll
    for (int off = 16; off; off >>= 1) {
        s += __shfl_xor(s, off, 32);
        d += __shfl_xor(d, off, 32);
    }
    if (lane == 0) { als[wave] = s; ald[wave] = d; }
}

// ---------- per-edge logits + segment max (self-loops appended implicitly) ----------
__global__ void edge_logits_k(const int* __restrict__ src, const int* __restrict__ dst,
                              int E, int N, int heads, const float* __restrict__ als,
                              const float* __restrict__ ald, float* __restrict__ logits,
                              unsigned* __restrict__ mOrd) {
    int ET = E + N;
    int total = ET * heads;
    int stride = gridDim.x * blockDim.x;
    for (int i = blockIdx.x * blockDim.x + threadIdx.x; i < total; i += stride) {
        int e = i / heads, hd = i - e * heads;
        int s = (e < E) ? src[e] : (e - E);
        int d = (e < E) ? dst[e] : (e - E);
        float l = als[s * heads + hd] + ald[d * heads + hd];
        l = (l >= 0.f) ? l : 0.2f * l;            // leaky_relu, slope 0.2
        logits[i] = l;
        atomicMax(mOrd + d * heads + hd, f2o(l));
    }
}

// ---------- wave-per-edge message aggregation (unnormalized), b128 gather ----------
__global__ void edge_agg_k(const int* __restrict__ src, const int* __restrict__ dst,
                           int E, int N, int heads, int C,
                           const float* __restrict__ logits, const unsigned* __restrict__ mOrd,
                           const float* __restrict__ h, float* __restrict__ denom,
                           float* __restrict__ agg) {
    int ET = E + N;
    int nw   = (gridDim.x * blockDim.x) >> 5;
    int wv   = (blockIdx.x * blockDim.x + threadIdx.x) >> 5;
    int lane = threadIdx.x & 31;
    for (int e = wv; e < ET; e += nw) {
        int s = (e < E) ? src[e] : (e - E);
        int d = (e < E) ? dst[e] : (e - E);
        const float* hp = h + (size_t)s * heads * C;
        float* ap = agg + (size_t)d * heads * C;
        for (int hd = 0; hd < heads; ++hd) {
            float ew = __expf(logits[(size_t)e * heads + hd] - o2f(mOrd[d * heads + hd]));
            if (lane == 0) atomicAdd(denom + d * heads + hd, ew);
            for (int c = lane * 4; c < C; c += 128) {      // C multiple of 128
                v4f hv = *(const v4f*)(hp + hd * C + c);   // global_load_b128
                float* a4 = ap + hd * C + c;
                atomicAdd(a4 + 0, ew * hv[0]);
                atomicAdd(a4 + 1, ew * hv[1]);
                atomicAdd(a4 + 2, ew * hv[2]);
                atomicAdd(a4 + 3, ew * hv[3]);
            }
        }
    }
}

// ---------- node-level normalization + bias (+ optional PReLU), float4 ----------
__global__ void finalize_k(const float* __restrict__ agg, const float* __restrict__ denom,
                           const float* __restrict__ bias, const float* __restrict__ prelu_w,
                           int do_prelu, float* __restrict__ out, int N, int heads, int C) {
    int  F4 = heads * C / 4;
    long total4 = (long)N * F4;
    long stride = (long)gridDim.x * blockDim.x;
    for (long i = (long)blockIdx.x * blockDim.x + threadIdx.x; i < total4; i += stride) {
        long n  = i / F4;
        int  q  = (int)(i - n * F4);
        int  c  = q * 4;
        int  hd = c / C;                 // a float4 never straddles a head boundary
        float inv = 1.f / denom[n * heads + hd];
        v4f v = ((const v4f*)agg)[i];
        v4f b = *(const v4f*)(bias + c);
        v4f r;
#pragma unroll
        for (int j = 0; j < 4; ++j) {
            float t = v[j] * inv + b[j];
            if (do_prelu) t = (t >= 0.f) ? t : prelu_w[0] * t;
            r[j] = t;
        }
        ((v4f*)out)[i] = r;
    }
}

extern "C" void kernel_launch(void* const* d_in, const int* in_sizes, int n_in,
                              void* d_out, int out_size, void* d_ws, size_t ws_size,
                              hipStream_t stream) {
    (void)in_sizes; (void)n_in; (void)out_size; (void)ws_size;
    const float* x     = (const float*)d_in[0];
    const int*   ei    = (const int*)d_in[1];
    const float* W1    = (const float*)d_in[2];
    const float* asrc1 = (const float*)d_in[3];
    const float* adst1 = (const float*)d_in[4];
    const float* b1    = (const float*)d_in[5];
    const float* prelu = (const float*)d_in[6];
    const float* W2    = (const float*)d_in[7];
    const float* asrc2 = (const float*)d_in[8];
    const float* adst2 = (const float*)d_in[9];
    const float* b2    = (const float*)d_in[10];
    float* out = (float*)d_out;

    constexpr int N = 30000, INC = 256, HID = 128, HEADS = 4, OUTC = 256;
    constexpr int E = 480000, ET = E + N, F1 = HEADS * HID;   // F1 = 512
    const int* srcA = ei;
    const int* dstA = ei + E;

    // ---- workspace layout (floats; every offset is a multiple of 4) ----
    float* ws = (float*)d_ws;
    size_t off = 0;
    auto wsa = [&](size_t nf) { float* p = ws + off; off += nf; return p; };
    float* bufA = wsa((size_t)N * F1);          // h1 linear, then h after PReLU
    float* bufB = wsa((size_t)N * F1);          // agg1; later z2 (first N*256) + agg2
    float* agg1 = bufB;
    float* z2   = bufB;
    float* agg2 = bufB + (size_t)N * OUTC;
    float*    als1 = wsa((size_t)N * HEADS);
    float*    ald1 = wsa((size_t)N * HEADS);
    unsigned* m1   = (unsigned*)wsa((size_t)N * HEADS);
    float*    den1 = wsa((size_t)N * HEADS);
    float*    als2 = wsa(N);
    float*    ald2 = wsa(N);
    unsigned* m2   = (unsigned*)wsa(N);
    float*    den2 = wsa(N);
    float*    lg1  = wsa((size_t)ET * HEADS);
    float*    lg2  = wsa(ET);

    const int BT = 256;
    auto nblk = [](long n) { return (int)((n + 255) / 256); };
    const unsigned NEG_INF_ORD = 0x007FFFFFu;   // f2o(-inf)

    // ======================= Layer 1 =======================
    {   // h1 = x @ W1  (fp32 WMMA, 16x64 strips)
        int groups = (N / 16) * (F1 / 64);
        wmma_gemm_f32_k<<<nblk((long)groups * 32), BT, 0, stream>>>(x, W1, bufA, N, INC, F1);
    }
    attn_coef_k<<<nblk((long)N * HEADS * 32), BT, 0, stream>>>(bufA, asrc1, adst1,
                                                               als1, ald1, N, HEADS, HID);
    fill_u32x4_k<<<nblk((long)N * HEADS / 4), BT, 0, stream>>>(m1, NEG_INF_ORD, (long)N * HEADS / 4);
    fill_u32x4_k<<<nblk((long)N * HEADS / 4), BT, 0, stream>>>((unsigned*)den1, 0u, (long)N * HEADS / 4);
    fill_u32x4_k<<<nblk((long)N * F1 / 4), BT, 0, stream>>>((unsigned*)agg1, 0u, (long)N * F1 / 4);
    edge_logits_k<<<nblk((long)ET * HEADS), BT, 0, stream>>>(srcA, dstA, E, N, HEADS,
                                                             als1, ald1, lg1, m1);
    edge_agg_k<<<nblk((long)ET * 32), BT, 0, stream>>>(srcA, dstA, E, N, HEADS, HID,
                                                       lg1, m1, bufA, den1, agg1);
    // h = PReLU(agg1/denom + b1)  -> overwrites bufA
    finalize_k<<<nblk((long)N * F1 / 4), BT, 0, stream>>>(agg1, den1, b1, prelu, 1,
                                                          bufA, N, HEADS, HID);

    // ======================= Layer 2 =======================
    {   // z2 = h @ W2  (fp32 WMMA); agg1 dead, reuse bufB
        int groups = (N / 16) * (OUTC / 64);
        wmma_gemm_f32_k<<<nblk((long)groups * 32), BT, 0, stream>>>(bufA, W2, z2, N, F1, OUTC);
    }
    attn_coef_k<<<nblk((long)N * 32), BT, 0, stream>>>(z2, asrc2, adst2, als2, ald2, N, 1, OUTC);
    fill_u32x4_k<<<nblk((long)N / 4), BT, 0, stream>>>(m2, NEG_INF_ORD, (long)N / 4);
    fill_u32x4_k<<<nblk((long)N / 4), BT, 0, stream>>>((unsigned*)den2, 0u, (long)N / 4);
    fill_u32x4_k<<<nblk((long)N * OUTC / 4), BT, 0, stream>>>((unsigned*)agg2, 0u, (long)N * OUTC / 4);
    edge_logits_k<<<nblk(ET), BT, 0, stream>>>(srcA, dstA, E, N, 1, als2, ald2, lg2, m2);
    edge_agg_k<<<nblk((long)ET * 32), BT, 0, stream>>>(srcA, dstA, E, N, 1, OUTC,
                                                       lg2, m2, z2, den2, agg2);
    // out = agg2/denom + b2  (1 head: mean == identity)
    finalize_k<<<nblk((long)N * OUTC / 4), BT, 0, stream>>>(agg2, den2, b2, prelu, 0,
                                                            out, N, 1, OUTC);
}